// RangeIndexedLinear_45380624449799
// MI455X (gfx1250) — compile-verified
//
#include <hip/hip_runtime.h>

typedef __attribute__((ext_vector_type(2))) float v2f;
typedef __attribute__((ext_vector_type(8))) float v8f;

#define BATCH        4096
#define IN_F         8192
#define OUT_F        8192
#define NG           1024
#define ROW_CHUNKS   8
#define ROWS_PER_CHUNK (BATCH / ROW_CHUNKS)   /* 512 */
#define COL_TILES    (IN_F / 16)              /* 512 */
#define WAVES_TOTAL  (COL_TILES * ROW_CHUNKS) /* 4096 */
#define WAVES_PER_BLOCK 4
#define K1_BLOCK     (WAVES_PER_BLOCK * 32)

// ---------------------------------------------------------------------------
// Kernel 1: column partial sums via WMMA (ones^T x Xtile accumulation).
// One wave handles one 16-column tile over a 512-row chunk.
// Per iteration: load a 4x16 f32 tile of x as the B matrix, A = all-ones.
// C += A*B  =>  every row of C holds the 16 running column sums.
// ---------------------------------------------------------------------------
__global__ void k_colsum_wmma(const float* __restrict__ x,
                              float* __restrict__ partials) {
    const int wave = blockIdx.x * WAVES_PER_BLOCK + (threadIdx.x >> 5);
    const int lane = threadIdx.x & 31;
    const int ct = wave % COL_TILES;       // 16-column tile index
    const int rc = wave / COL_TILES;       // row chunk index
    const int n  = lane & 15;              // column within tile
    const int kh = lane >> 4;              // K half (rows +0/+1 vs +2/+3)

    const float* p = x + (size_t)(rc * ROWS_PER_CHUNK + 2 * kh) * IN_F
                       + (size_t)ct * 16 + n;

    v2f a; a.x = 1.0f; a.y = 1.0f;         // all-ones A matrix (16x4)
    v8f c = {};

#pragma unroll 4
    for (int r = 0; r < ROWS_PER_CHUNK; r += 4) {
        v2f b;
        b.x = p[0];                        // row (base + 2*kh)
        b.y = p[IN_F];                     // row (base + 2*kh + 1)
        c = __builtin_amdgcn_wmma_f32_16x16x4_f32(
                false, a, false, b, (short)0, c, false, false);
        p += 4 * (size_t)IN_F;
    }

    // C/D layout: VGPR0 lanes0-15 = (M=0, N=lane). All M rows identical.
    if (lane < 16) {
        partials[(size_t)rc * IN_F + (size_t)ct * 16 + n] = c[0];
    }
}

// ---------------------------------------------------------------------------
// Kernel 2: finalize per-column value, bucket lookup, weight gather, term.
// ---------------------------------------------------------------------------
__global__ void k_terms(const float* __restrict__ partials,
                        const float* __restrict__ mins,
                        const float* __restrict__ maxs,
                        const int*   __restrict__ start_pos,
                        const int*   __restrict__ offsets,
                        const int*   __restrict__ sizes,
                        const float* __restrict__ wflat,
                        float*       __restrict__ terms) {
    __shared__ float smins[NG];
    for (int i = threadIdx.x; i < NG; i += blockDim.x) smins[i] = mins[i];
    __syncthreads();

    const int j = blockIdx.x * blockDim.x + threadIdx.x;
    if (j >= IN_F) return;

    float cs = 0.0f;
    for (int rc = 0; rc < ROW_CHUNKS; ++rc)
        cs += partials[(size_t)rc * IN_F + j];
    const float val = cs * (1.0f / (float)BATCH);

    // searchsorted(mins, val, side='right') - 1
    int lo = 0, hi = NG;
    while (lo < hi) {
        int mid = (lo + hi) >> 1;
        if (smins[mid] <= val) lo = mid + 1; else hi = mid;
    }
    const int g  = lo - 1;
    const int gc = g < 0 ? 0 : (g > NG - 1 ? NG - 1 : g);

    const bool hit = (g >= 0) && (val >= smins[gc]) && (val <= maxs[gc]);
    const int  pos = j - start_pos[gc];
    const bool pos_ok = (pos >= 0) && (pos < sizes[gc]);

    float w = 0.0f;
    if (hit && pos_ok) {
        long long idx  = (long long)offsets[gc] + (long long)pos;
        const long long maxi = (long long)NG * IN_F - 1;
        idx = idx < 0 ? 0 : (idx > maxi ? maxi : idx);
        w = wflat[idx];
    }
    terms[j] = val * w;
}

// ---------------------------------------------------------------------------
// Kernel 3: deterministic single-block reduction of 8192 terms -> scalar s.
// ---------------------------------------------------------------------------
__global__ void k_reduce(const float* __restrict__ terms,
                         float* __restrict__ sOut) {
    __shared__ float sh[256];
    float acc = 0.0f;
    for (int j = threadIdx.x; j < IN_F; j += 256) acc += terms[j];
    sh[threadIdx.x] = acc;
    __syncthreads();
    for (int st = 128; st > 0; st >>= 1) {
        if (threadIdx.x < st) sh[threadIdx.x] += sh[threadIdx.x + st];
        __syncthreads();
    }
    if (threadIdx.x == 0) sOut[0] = sh[0];
}

// ---------------------------------------------------------------------------
// Kernel 4: fill output. Row 0 = mask ? s : 0, everything else zero.
// float4 (B128) stores; 128 MiB streaming write.
// ---------------------------------------------------------------------------
__global__ void k_fill(float* __restrict__ out,
                       const int* __restrict__ mask,
                       const float* __restrict__ sPtr) {
    const size_t i4 = (size_t)blockIdx.x * blockDim.x + threadIdx.x;
    float4 v = make_float4(0.0f, 0.0f, 0.0f, 0.0f);
    if (i4 < (size_t)(OUT_F / 4)) {               // row 0 region
        const float s = sPtr[0];
        const int b = (int)i4 * 4;
        v.x = mask[b + 0] ? s : 0.0f;
        v.y = mask[b + 1] ? s : 0.0f;
        v.z = mask[b + 2] ? s : 0.0f;
        v.w = mask[b + 3] ? s : 0.0f;
    }
    reinterpret_cast<float4*>(out)[i4] = v;
}

extern "C" void kernel_launch(void* const* d_in, const int* in_sizes, int n_in,
                              void* d_out, int out_size, void* d_ws, size_t ws_size,
                              hipStream_t stream) {
    const float* x         = (const float*)d_in[0];
    const float* wflat     = (const float*)d_in[1];
    const float* mins      = (const float*)d_in[2];
    const float* maxs      = (const float*)d_in[3];
    const int*   start_pos = (const int*)  d_in[4];
    const int*   offsets   = (const int*)  d_in[5];
    const int*   sizes     = (const int*)  d_in[6];
    const int*   mask      = (const int*)  d_in[7];
    float*       out       = (float*)d_out;

    float* wsf      = (float*)d_ws;
    float* partials = wsf;                         // ROW_CHUNKS * IN_F floats
    float* terms    = wsf + ROW_CHUNKS * IN_F;     // IN_F floats
    float* sv       = terms + IN_F;                // 1 float

    k_colsum_wmma<<<WAVES_TOTAL / WAVES_PER_BLOCK, K1_BLOCK, 0, stream>>>(x, partials);
    k_terms<<<IN_F / 256, 256, 0, stream>>>(partials, mins, maxs,
                                            start_pos, offsets, sizes, wflat, terms);
    k_reduce<<<1, 256, 0, stream>>>(terms, sv);

    const size_t n4 = (size_t)BATCH * OUT_F / 4;   // 8,388,608 float4 stores
    k_fill<<<(unsigned)((n4 + 255) / 256), 256, 0, stream>>>(out, mask, sv);
}